// SwigluMoEBlock_23098334118516
// MI455X (gfx1250) — compile-verified
//
#include <hip/hip_runtime.h>
#include <math.h>

typedef __attribute__((ext_vector_type(2))) float v2f;
typedef __attribute__((ext_vector_type(8))) float v8f;

static constexpr int TOKENS = 8192;   // B*S = 4*2048
static constexpr int HDIM   = 2048;   // hidden (K of GEMM1, N of GEMM2)
static constexpr int NH2    = 4096;   // 2*I   (N of GEMM1)
static constexpr int IDIM   = 2048;   // I     (K of GEMM2)
static constexpr int NEXP   = 8;

static constexpr int BM   = 64;       // token tile
static constexpr int BN   = 128;      // output-col tile (2 N-waves x 64)
static constexpr int BK   = 32;       // K tile
static constexpr int LDSS = 36;       // LDS row stride (floats): 16B aligned, conflict-free
static constexpr int ASZ  = BM * LDSS;   // 2304 floats
static constexpr int BSZ  = BN * LDSS;   // 4608 floats
static constexpr int SMEM_FLOATS = 2 * (ASZ + BSZ);  // 13824 floats = 55296 B

// ---------------------------------------------------------------------------
// CDNA5 async global->LDS copy (ASYNCcnt path), with safe fallback
// ---------------------------------------------------------------------------
#if defined(__HIP_DEVICE_COMPILE__) && __has_builtin(__builtin_amdgcn_global_load_async_to_lds_b128)
#define USE_ASYNC_COPY 1
#else
#define USE_ASYNC_COPY 0
#endif

#if USE_ASYNC_COPY
typedef int b128_t __attribute__((vector_size(16)));
typedef __attribute__((address_space(1))) b128_t* as1_b128;
typedef __attribute__((address_space(3))) b128_t* as3_b128;
__device__ __forceinline__ void async_b128(const float* g, float* l) {
  __builtin_amdgcn_global_load_async_to_lds_b128(
      (as1_b128)(b128_t*)g, (as3_b128)(b128_t*)l, 0, 0);
}
#if __has_builtin(__builtin_amdgcn_s_wait_asynccnt)
#define WAIT_ASYNC() __builtin_amdgcn_s_wait_asynccnt(0)
#else
#define WAIT_ASYNC() asm volatile("s_wait_asynccnt 0x0" ::: "memory")
#endif
#else
#define WAIT_ASYNC() do {} while (0)
#endif

// Copy one BK-slice of the A (64xBK) and B (128xBK) tiles.
// ag/bg: this thread's global row base + fixed column offset; sAd/sBd: its LDS slots.
__device__ __forceinline__ void copy_tile(const float* ag, const float* bg, int kk,
                                          float* sAd, float* sBd) {
#if USE_ASYNC_COPY
  async_b128(ag + kk,      sAd);
  async_b128(ag + kk + 4,  sAd + 4);
  async_b128(bg + kk,      sBd);
  async_b128(bg + kk + 4,  sBd + 4);
  async_b128(bg + kk + 8,  sBd + 8);
  async_b128(bg + kk + 12, sBd + 12);
#else
  const float4 a0 = *(const float4*)(ag + kk);
  const float4 a1 = *(const float4*)(ag + kk + 4);
  *(float4*)(sAd)     = a0;
  *(float4*)(sAd + 4) = a1;
  const float4 b0 = *(const float4*)(bg + kk);
  const float4 b1 = *(const float4*)(bg + kk + 4);
  const float4 b2 = *(const float4*)(bg + kk + 8);
  const float4 b3 = *(const float4*)(bg + kk + 12);
  *(float4*)(sBd)      = b0;
  *(float4*)(sBd + 4)  = b1;
  *(float4*)(sBd + 8)  = b2;
  *(float4*)(sBd + 12) = b3;
#endif
}

// One BK step of WMMA: wave computes a 16(M) x 64(N) strip = 4 C tiles.
__device__ __forceinline__ void mma_tile(const float* sA, const float* sB,
                                         int lane, int wm, int wn, v8f c[4]) {
  const int ar = (wm + (lane & 15)) * LDSS;
  const int kb = (lane >> 4) * 2;          // lanes 16-31 hold K={2,3}
#pragma unroll
  for (int k4 = 0; k4 < BK; k4 += 4) {
    const int kc = k4 + kb;
    const v2f a = *(const v2f*)&sA[ar + kc];
#pragma unroll
    for (int j = 0; j < 4; ++j) {
      const v2f b = *(const v2f*)&sB[(wn + j * 16 + (lane & 15)) * LDSS + kc];
      c[j] = __builtin_amdgcn_wmma_f32_16x16x4_f32(false, a, false, b, (short)0,
                                                   c[j], false, false);
    }
  }
}

// ---------------------------------------------------------------------------
// Zero output + per-expert counters
// ---------------------------------------------------------------------------
__global__ __launch_bounds__(256)
void zero_kernel(float* __restrict__ out, size_t n, int* __restrict__ cnt) {
  size_t i = (size_t)blockIdx.x * blockDim.x + threadIdx.x;
  size_t stride = (size_t)gridDim.x * blockDim.x;
  for (; i < n; i += stride) out[i] = 0.0f;
  if (blockIdx.x == 0 && threadIdx.x < NEXP) cnt[threadIdx.x] = 0;
}

// ---------------------------------------------------------------------------
// Gate: one wave32 per token; butterfly reduce; exact top-2 + softmax;
// atomic append to per-expert token lists.
// ---------------------------------------------------------------------------
__global__ __launch_bounds__(256)
void gate_kernel(const float* __restrict__ x,
                 const float* __restrict__ gw,
                 const float* __restrict__ gb,
                 int*   __restrict__ cnt,
                 int*   __restrict__ idx,
                 float* __restrict__ wgt) {
  const int lane = threadIdx.x & 31;
  const int wv   = threadIdx.x >> 5;
  const int t    = blockIdx.x * 8 + wv;
  if (t >= TOKENS) return;

  const float* xr = x + (size_t)t * HDIM;
  float acc[NEXP];
#pragma unroll
  for (int e = 0; e < NEXP; ++e) acc[e] = 0.0f;

  for (int h = lane; h < HDIM; h += 32) {
    const float xv = xr[h];
#pragma unroll
    for (int e = 0; e < NEXP; ++e) acc[e] += xv * gw[e * HDIM + h];
  }
#pragma unroll
  for (int off = 16; off > 0; off >>= 1) {
#pragma unroll
    for (int e = 0; e < NEXP; ++e) acc[e] += __shfl_xor(acc[e], off, 32);
  }

  if (lane == 0) {
    float best = -3.0e38f, second = -3.0e38f;
    int bi = 0, si = 0;
#pragma unroll
    for (int e = 0; e < NEXP; ++e) {
      const float v = acc[e] + gb[e];
      if (v > best)        { second = best; si = bi; best = v; bi = e; }
      else if (v > second) { second = v; si = e; }
    }
    const float w0 = 1.0f / (1.0f + __expf(second - best));
    const float w1 = 1.0f - w0;
    const int p0 = atomicAdd(&cnt[bi], 1);
    idx[bi * TOKENS + p0] = t;  wgt[bi * TOKENS + p0] = w0;
    const int p1 = atomicAdd(&cnt[si], 1);
    idx[si * TOKENS + p1] = t;  wgt[si * TOKENS + p1] = w1;
  }
}

// ---------------------------------------------------------------------------
// GEMM1: h = gather(x) @ w1[e]^T + b1[e]; interleaved SwiGLU -> y
// ---------------------------------------------------------------------------
__global__ __launch_bounds__(256)
void expert_gemm1(const float* __restrict__ x,
                  const float* __restrict__ w1e,
                  const float* __restrict__ b1e,
                  const int*   __restrict__ cnt,
                  const int*   __restrict__ idx,
                  float*       __restrict__ ybuf,
                  int e) {
  const int count = cnt[e];
  const int m0 = blockIdx.y * BM;
  if (m0 >= count) return;                 // uniform exit: EXEC all-ones at WMMA
  const int n0 = blockIdx.x * BN;

  __shared__ __attribute__((aligned(16))) float smem[SMEM_FLOATS];
  // layout: A0 [0,ASZ) | A1 [ASZ,2*ASZ) | B0 [2*ASZ, 2*ASZ+BSZ) | B1 [...]

  const int tid  = threadIdx.x;
  const int lane = tid & 31;
  const int wv   = tid >> 5;
  const int wm   = (wv & 3) * 16;          // wave M offset
  const int wn   = (wv >> 2) * 64;         // wave N offset (16x64 strip)

  // cooperative-copy coordinates
  const int arow = tid >> 2, acol = (tid & 3) * 8;   // A: 64 x BK, 8 floats/thread
  const int brow = tid >> 1, bcol = (tid & 1) * 16;  // B: 128 x BK, 16 floats/thread

  const int xrow = m0 + arow;
  const int tok  = (xrow < count) ? idx[xrow] : 0;   // clamp: finite garbage, discarded
  const float* ag = x + (size_t)tok * HDIM + acol;
  const float* bg = w1e + (size_t)(n0 + brow) * HDIM + bcol;
  const int aoff = arow * LDSS + acol;
  const int boff = brow * LDSS + bcol;

  v8f c[4] = {v8f{}, v8f{}, v8f{}, v8f{}};

  copy_tile(ag, bg, 0, smem + aoff, smem + 2 * ASZ + boff);
  WAIT_ASYNC();
  __syncthreads();

  int cur = 0;
  for (int kk = BK; kk < HDIM; kk += BK) {
    const int nxt = cur ^ 1;
    copy_tile(ag, bg, kk, smem + nxt * ASZ + aoff, smem + 2 * ASZ + nxt * BSZ + boff);
    mma_tile(smem + cur * ASZ, smem + 2 * ASZ + cur * BSZ, lane, wm, wn, c);
    WAIT_ASYNC();
    __syncthreads();
    cur = nxt;
  }
  mma_tile(smem + cur * ASZ, smem + 2 * ASZ + cur * BSZ, lane, wm, wn, c);
  __syncthreads();                          // before reusing smem as staging

  // Epilogue: bias + interleaved SwiGLU (pair via shfl_xor(1)), stage in LDS,
  // then coalesced float4 stores of the 64 x 64 y tile.
  float* stage = smem;                      // 64 x 68
  const int  hi8  = (lane >> 4) << 3;
  const bool even = ((lane & 1) == 0);
#pragma unroll
  for (int j = 0; j < 4; ++j) {
    const int ncol = n0 + wn + j * 16 + (lane & 15);
    const float bias = b1e[ncol];
#pragma unroll
    for (int r = 0; r < 8; ++r) {
      const float hv  = c[j][r] + bias;
      const float pv  = __shfl_xor(hv, 1, 32);
      const float glu = even ? hv : pv;
      const float lin = even ? pv : hv;
      const float yv  = glu * (1.0f / (1.0f + __expf(-1.702f * glu))) * (lin + 1.0f);
      if (even) stage[(wm + hi8 + r) * 68 + ((wn + j * 16 + (lane & 15)) >> 1)] = yv;
    }
  }
  __syncthreads();

  const int srow = tid >> 2;
  const int scol = (tid & 3) * 16;
  float* ydst = ybuf + (size_t)(m0 + srow) * IDIM + (n0 >> 1) + scol;
#pragma unroll
  for (int q = 0; q < 4; ++q) {
    *(float4*)(ydst + q * 4) = *(const float4*)&stage[srow * 68 + scol + q * 4];
  }
}

// ---------------------------------------------------------------------------
// GEMM2: out[tok(m)] += wgt(m) * (y @ w2[e]^T + b2[e])
// Experts are stream-serialized and each token appears once per expert list,
// so plain read-modify-write (no atomics) is race-free and deterministic.
// ---------------------------------------------------------------------------
__global__ __launch_bounds__(256)
void expert_gemm2(const float* __restrict__ ybuf,
                  const float* __restrict__ w2e,
                  const float* __restrict__ b2e,
                  const int*   __restrict__ cnt,
                  const int*   __restrict__ idx,
                  const float* __restrict__ wgt,
                  float*       __restrict__ out,
                  int e) {
  const int count = cnt[e];
  const int m0 = blockIdx.y * BM;
  if (m0 >= count) return;
  const int n0 = blockIdx.x * BN;

  __shared__ __attribute__((aligned(16))) float smem[SMEM_FLOATS];

  const int tid  = threadIdx.x;
  const int lane = tid & 31;
  const int wv   = tid >> 5;
  const int wm   = (wv & 3) * 16;
  const int wn   = (wv >> 2) * 64;

  const int arow = tid >> 2, acol = (tid & 3) * 8;
  const int brow = tid >> 1, bcol = (tid & 1) * 16;

  const float* ag = ybuf + (size_t)(m0 + arow) * IDIM + acol;  // list-ordered rows
  const float* bg = w2e + (size_t)(n0 + brow) * IDIM + bcol;
  const int aoff = arow * LDSS + acol;
  const int boff = brow * LDSS + bcol;

  v8f c[4] = {v8f{}, v8f{}, v8f{}, v8f{}};

  copy_tile(ag, bg, 0, smem + aoff, smem + 2 * ASZ + boff);
  WAIT_ASYNC();
  __syncthreads();

  int cur = 0;
  for (int kk = BK; kk < IDIM; kk += BK) {
    const int nxt = cur ^ 1;
    copy_tile(ag, bg, kk, smem + nxt * ASZ + aoff, smem + 2 * ASZ + nxt * BSZ + boff);
    mma_tile(smem + cur * ASZ, smem + 2 * ASZ + cur * BSZ, lane, wm, wn, c);
    WAIT_ASYNC();
    __syncthreads();
    cur = nxt;
  }
  mma_tile(smem + cur * ASZ, smem + 2 * ASZ + cur * BSZ, lane, wm, wn, c);
  __syncthreads();

  // Epilogue: bias + routing weight, stage full 64 x 128 tile, coalesced
  // float4 read-modify-write into out rows.
  float* stage = smem;                      // 64 x 132
  const int hi8 = (lane >> 4) << 3;
  float rowW[8];
#pragma unroll
  for (int r = 0; r < 8; ++r) {
    const int row = m0 + wm + hi8 + r;
    rowW[r] = (row < count) ? wgt[row] : 0.0f;
  }
#pragma unroll
  for (int j = 0; j < 4; ++j) {
    const int ncol = n0 + wn + j * 16 + (lane & 15);
    const float bias = b2e[ncol];
#pragma unroll
    for (int r = 0; r < 8; ++r) {
      stage[(wm + hi8 + r) * 132 + wn + j * 16 + (lane & 15)] = (c[j][r] + bias) * rowW[r];
    }
  }
  __syncthreads();

  const int srow = tid >> 2;              // 0..63
  const int scol = (tid & 3) * 32;        // 32 floats per thread
  const int grow = m0 + srow;
  if (grow < count) {
    const int tok = idx[grow];
    float* orow = out + (size_t)tok * HDIM + n0 + scol;
#pragma unroll
    for (int q = 0; q < 8; ++q) {
      float4 ov = *(float4*)(orow + q * 4);
      const float4 sv = *(const float4*)&stage[srow * 132 + scol + q * 4];
      ov.x += sv.x; ov.y += sv.y; ov.z += sv.z; ov.w += sv.w;
      *(float4*)(orow + q * 4) = ov;
    }
  }
}

// ---------------------------------------------------------------------------
// Host launcher
// ---------------------------------------------------------------------------
extern "C" void kernel_launch(void* const* d_in, const int* in_sizes, int n_in,
                              void* d_out, int out_size, void* d_ws, size_t ws_size,
                              hipStream_t stream) {
  const float* x  = (const float*)d_in[0];   // (4,2048,2048)
  const float* gw = (const float*)d_in[1];   // (8,2048)
  const float* gb = (const float*)d_in[2];   // (8,)
  const float* w1 = (const float*)d_in[3];   // (8,4096,2048)
  const float* b1 = (const float*)d_in[4];   // (8,4096)
  const float* w2 = (const float*)d_in[5];   // (8,2048,2048)
  const float* b2 = (const float*)d_in[6];   // (8,2048)
  float* out = (float*)d_out;                // (4,2048,2048)

  char*  ws   = (char*)d_ws;
  int*   cnt  = (int*)ws;
  int*   idx  = (int*)(ws + 256);
  float* wgt  = (float*)(ws + 256 + (size_t)NEXP * TOKENS * 4);
  float* ybuf = (float*)(ws + 256 + 2 * (size_t)NEXP * TOKENS * 4);  // 64 MB

  zero_kernel<<<2048, 256, 0, stream>>>(out, (size_t)TOKENS * HDIM, cnt);
  gate_kernel<<<TOKENS / 8, 256, 0, stream>>>(x, gw, gb, cnt, idx, wgt);

  for (int e = 0; e < NEXP; ++e) {
    expert_gemm1<<<dim3(NH2 / BN, TOKENS / BM), 256, 0, stream>>>(
        x, w1 + (size_t)e * NH2 * HDIM, b1 + (size_t)e * NH2,
        cnt, idx + (size_t)e * TOKENS, ybuf, e);
    expert_gemm2<<<dim3(HDIM / BN, TOKENS / BM), 256, 0, stream>>>(
        ybuf, w2 + (size_t)e * HDIM * IDIM, b2 + (size_t)e * HDIM,
        cnt, idx + (size_t)e * TOKENS, wgt + (size_t)e * TOKENS, out, e);
  }
}